// BilinearUpSampling3D_17600775979834
// MI455X (gfx1250) — compile-verified
//
#include <hip/hip_runtime.h>
#include <cstdint>

// ---------------------------------------------------------------------------
// 2x trilinear upsample (TF-v1 asymmetric coords, scale exactly 0.5 per axis):
//   out[2i]   = in[i]
//   out[2i+1] = 0.5*(in[i] + in[min(i+1, n-1)])
// Separable product over H, W, D. Pure bandwidth problem (~44us HBM floor:
// 113 MB read + 906 MB NT write @ 23.3 TB/s).
// Strategy: async global->LDS tile staging (ASYNCcnt path, b128 per lane),
// per-thread 2x2x2 cells from 8 ds_load_b128 reads, NT b128 streaming stores.
// Loops fully unrolled for MLP (16 stores / 4 async loads in flight).
// ---------------------------------------------------------------------------

typedef __attribute__((ext_vector_type(4))) float v4f;
typedef __attribute__((ext_vector_type(4))) int   v4i;

#define AS1 __attribute__((address_space(1)))
#define AS3 __attribute__((address_space(3)))

#if defined(__gfx1250__) && __has_builtin(__builtin_amdgcn_global_load_async_to_lds_b128)
#define HAVE_ASYNC_LDS 1
#else
#define HAVE_ASYNC_LDS 0
#endif

constexpr int B_ = 2, H_ = 96, W_ = 96, D_ = 48, C_ = 32;
constexpr int TH_ = 192, TW_ = 192, TD_ = 96;
constexpr int OT = 8;                 // output tile edge (spatial)
constexpr int IT = OT / 2 + 1;        // 5: input tile edge (with +1 halo)
constexpr int NSP   = IT * IT * IT;        // 125 input points per tile
constexpr int NLOAD = NSP * (C_ / 4);      // 1000 float4 slots to stage
constexpr int NCELL = (OT/2)*(OT/2)*(OT/2);// 64 cells per tile
constexpr int NTASK = NCELL * (C_ / 4);    // 512 (cell, c4) tasks

__global__ __launch_bounds__(256)
void upsample3d_2x_kernel(const float* __restrict__ x, float* __restrict__ out) {
    __shared__ float lds[NSP * C_];   // 4000 floats = 16 KB

    const int bd  = blockIdx.x % (TD_ / OT);
    const int b   = blockIdx.x / (TD_ / OT);
    const int bw  = blockIdx.y;
    const int bh  = blockIdx.z;
    const int tid = threadIdx.x;

    const int ih0 = bh * (OT / 2);    // input tile origin
    const int iw0 = bw * (OT / 2);
    const int id0 = bd * (OT / 2);

    // ---- stage clamped 5x5x5x32 input tile into LDS (async b128 per lane) ----
#pragma unroll
    for (int it = 0; it < 4; ++it) {          // 4*256 >= NLOAD, last iter guarded
        const int l = tid + it * 256;
        if (it < 3 || l < NLOAD) {
            const int c4 = l & 7;
            const int s  = l >> 3;
            const int k  = s % IT;
            const int j  = (s / IT) % IT;
            const int i  = s / (IT * IT);
            const int gh = min(ih0 + i, H_ - 1);
            const int gw = min(iw0 + j, W_ - 1);
            const int gd = min(id0 + k, D_ - 1);
            const size_t gidx = ((((size_t)b * H_ + gh) * W_ + gw) * D_ + gd) * C_ + c4 * 4;
#if HAVE_ASYNC_LDS
            __builtin_amdgcn_global_load_async_to_lds_b128(
                (AS1 v4i*)(x + gidx), (AS3 v4i*)(&lds[l * 4]), 0, 0);
#else
            *(v4f*)(&lds[l * 4]) = *(const v4f*)(x + gidx);
#endif
        }
    }
#if HAVE_ASYNC_LDS
#if __has_builtin(__builtin_amdgcn_s_wait_asynccnt)
    __builtin_amdgcn_s_wait_asynccnt(0);
#else
    asm volatile("s_wait_asynccnt 0" ::: "memory");
#endif
#endif
    __syncthreads();

    // ---- each task: one 2x2x2 output cell for one float4 channel group ----
    const v4f* lds4 = (const v4f*)lds;
#pragma unroll
    for (int it = 0; it < NTASK / 256; ++it) {    // exactly 2 iterations
        const int q    = tid + it * 256;
        const int c4   = q & 7;
        const int cell = q >> 3;
        const int ck = cell & 3;
        const int cj = (cell >> 2) & 3;
        const int ci = cell >> 4;

#define LSLOT(i, j, k) (((((i) * IT) + (j)) * IT + (k)) * (C_ / 4) + c4)
        const v4f v000 = lds4[LSLOT(ci,     cj,     ck    )];
        const v4f v001 = lds4[LSLOT(ci,     cj,     ck + 1)];
        const v4f v010 = lds4[LSLOT(ci,     cj + 1, ck    )];
        const v4f v011 = lds4[LSLOT(ci,     cj + 1, ck + 1)];
        const v4f v100 = lds4[LSLOT(ci + 1, cj,     ck    )];
        const v4f v101 = lds4[LSLOT(ci + 1, cj,     ck + 1)];
        const v4f v110 = lds4[LSLOT(ci + 1, cj + 1, ck    )];
        const v4f v111 = lds4[LSLOT(ci + 1, cj + 1, ck + 1)];
#undef LSLOT

        const v4f o000 = v000;
        const v4f o001 = (v000 + v001) * 0.5f;
        const v4f o010 = (v000 + v010) * 0.5f;
        const v4f o100 = (v000 + v100) * 0.5f;
        const v4f s_wd = (v000 + v001) + (v010 + v011);   // d & w neighbors
        const v4f o011 = s_wd * 0.25f;
        const v4f o101 = ((v000 + v001) + (v100 + v101)) * 0.25f;
        const v4f o110 = ((v000 + v010) + (v100 + v110)) * 0.25f;
        const v4f o111 = (s_wd + ((v100 + v101) + (v110 + v111))) * 0.125f;

        const int oh = bh * OT + 2 * ci;
        const int ow = bw * OT + 2 * cj;
        const int od = bd * OT + 2 * ck;
        float* p = out + ((((size_t)b * TH_ + oh) * TW_ + ow) * TD_ + od) * C_ + c4 * 4;
        constexpr size_t SH = (size_t)TW_ * TD_ * C_;  // stride for oh+1
        constexpr size_t SW = (size_t)TD_ * C_;        // stride for ow+1
        constexpr size_t SD = C_;                      // stride for od+1

        __builtin_nontemporal_store(o000, (v4f*)(p));
        __builtin_nontemporal_store(o001, (v4f*)(p + SD));
        __builtin_nontemporal_store(o010, (v4f*)(p + SW));
        __builtin_nontemporal_store(o011, (v4f*)(p + SW + SD));
        __builtin_nontemporal_store(o100, (v4f*)(p + SH));
        __builtin_nontemporal_store(o101, (v4f*)(p + SH + SD));
        __builtin_nontemporal_store(o110, (v4f*)(p + SH + SW));
        __builtin_nontemporal_store(o111, (v4f*)(p + SH + SW + SD));
    }
}

extern "C" void kernel_launch(void* const* d_in, const int* in_sizes, int n_in,
                              void* d_out, int out_size, void* d_ws, size_t ws_size,
                              hipStream_t stream) {
    (void)in_sizes; (void)n_in; (void)out_size; (void)d_ws; (void)ws_size;
    const float* x = (const float*)d_in[0];
    float* out = (float*)d_out;
    dim3 grid((TD_ / OT) * B_,  // 24: d-tiles x batch
              TW_ / OT,         // 24
              TH_ / OT);        // 24
    upsample3d_2x_kernel<<<grid, dim3(256), 0, stream>>>(x, out);
}